// ColorICP_33320356282475
// MI455X (gfx1250) — compile-verified
//
#include <hip/hip_runtime.h>
#include <math.h>

typedef __attribute__((ext_vector_type(2))) float v2f;
typedef __attribute__((ext_vector_type(8))) float v8f;

#define H_IMG 960
#define W_IMG 1280
#define NPIX (H_IMG * W_IMG)
#define HUBER_BC 0.02f
#define DAMPING_C 0.001f
#define DIST_THRESH_C 0.1f
#define SQRT_LAMBDA_RGB 1.0e-3f

// ws float-indexed layout
#define WS_M 0        // 49 floats: 7x7 augmented normal-equation accumulator
#define WS_POSE 64    // 16 floats: current 4x4 pose, row-major
#define WS_MINMAX 96  // 2 ints: [0]=min bits, [1]=max bits of depth1
#define WS_VN 1024    // optional: NPIX*8 floats {vx,vy,vz,_,nx,ny,nz,_}
#define WS_VN_BYTES ((size_t)4 * (WS_VN + (size_t)NPIX * 8))

struct F3 { float x, y, z; };
__device__ __forceinline__ F3 f3(float a, float b, float c) { F3 r; r.x=a; r.y=b; r.z=c; return r; }
__device__ __forceinline__ F3 add3(F3 a, F3 b) { return f3(a.x+b.x, a.y+b.y, a.z+b.z); }
__device__ __forceinline__ F3 sub3(F3 a, F3 b) { return f3(a.x-b.x, a.y-b.y, a.z-b.z); }
__device__ __forceinline__ F3 scl3(F3 a, float s) { return f3(a.x*s, a.y*s, a.z*s); }
__device__ __forceinline__ float dot3(F3 a, F3 b) { return a.x*b.x + a.y*b.y + a.z*b.z; }
__device__ __forceinline__ F3 cross3(F3 a, F3 b) {
  return f3(a.y*b.z - a.z*b.y, a.z*b.x - a.x*b.z, a.x*b.y - a.y*b.x);
}
__device__ __forceinline__ int imin2(int a, int b) { return a < b ? a : b; }
__device__ __forceinline__ int imax2(int a, int b) { return a > b ? a : b; }

__device__ __forceinline__ F3 vert1(const float* __restrict__ d1, int y, int x,
                                    float fx, float fy, float cx, float cy) {
  float d = d1[y * W_IMG + x];
  return f3(((float)x - cx) / fx * d, ((float)y - cy) / fy * d, d);
}

// compute_normal(vertex1) at pixel (y,x): unnormalized sobel of vertex map,
// cross(dx,dy), normalize, zero where depth1 equals global min/max.
__device__ F3 normal1At(const float* __restrict__ d1, int y, int x,
                        float fx, float fy, float cx, float cy,
                        float dmin, float dmax) {
  int xm = imax2(x - 1, 0), xp = imin2(x + 1, W_IMG - 1);
  int ym = imax2(y - 1, 0), yp = imin2(y + 1, H_IMG - 1);
  F3 vmm = vert1(d1, ym, xm, fx, fy, cx, cy);
  F3 vm0 = vert1(d1, ym, x,  fx, fy, cx, cy);
  F3 vmp = vert1(d1, ym, xp, fx, fy, cx, cy);
  F3 v0m = vert1(d1, y,  xm, fx, fy, cx, cy);
  F3 v0p = vert1(d1, y,  xp, fx, fy, cx, cy);
  F3 vpm = vert1(d1, yp, xm, fx, fy, cx, cy);
  F3 vp0 = vert1(d1, yp, x,  fx, fy, cx, cy);
  F3 vpp = vert1(d1, yp, xp, fx, fy, cx, cy);
  F3 dx = add3(add3(sub3(vmp, vmm), scl3(sub3(v0p, v0m), 2.0f)), sub3(vpp, vpm));
  F3 dy = add3(add3(sub3(vpm, vmm), scl3(sub3(vp0, vm0), 2.0f)), sub3(vpp, vmp));
  F3 n = cross3(dx, dy);
  float nn = sqrtf(dot3(n, n));
  n = scl3(n, 1.0f / (nn + 1e-8f));
  float d = d1[y * W_IMG + x];
  bool invalid = (d <= dmin) || (d >= dmax);
  return invalid ? f3(0.f, 0.f, 0.f) : n;
}

__global__ void init_kernel(const float* __restrict__ pose10, float* __restrict__ ws_f) {
  int t = threadIdx.x;
  if (t < 49) ws_f[WS_M + t] = 0.0f;
  if (t < 16) ws_f[WS_POSE + t] = pose10[t];
  if (t == 0) {
    int* mm = (int*)(ws_f + WS_MINMAX);
    mm[0] = 0x7F7FFFFF; // FLT_MAX bits (depth1 > 0 so int-bit compare == float compare)
    mm[1] = 0;          // 0.0f bits
  }
}

__global__ void minmax_kernel(const float* __restrict__ d1, float* __restrict__ ws_f) {
  __shared__ int smin[256];
  __shared__ int smax[256];
  int t = threadIdx.x;
  float mn = 3.0e38f, mx = 0.0f;
  for (int i = blockIdx.x * blockDim.x + t; i < NPIX; i += gridDim.x * blockDim.x) {
    float d = d1[i];
    mn = fminf(mn, d);
    mx = fmaxf(mx, d);
  }
  smin[t] = __float_as_int(mn);
  smax[t] = __float_as_int(mx);
  __syncthreads();
  for (int s = 128; s > 0; s >>= 1) {
    if (t < s) {
      smin[t] = imin2(smin[t], smin[t + s]);
      smax[t] = imax2(smax[t], smax[t + s]);
    }
    __syncthreads();
  }
  if (t == 0) {
    int* mm = (int*)(ws_f + WS_MINMAX);
    atomicMin(&mm[0], smin[0]);
    atomicMax(&mm[1], smax[0]);
  }
}

// Streaming precompute of vertex1 + normal1 into ws (fast path, one pass).
__global__ void precompute_vn_kernel(const float* __restrict__ depth1,
                                     const float* __restrict__ Kp,
                                     float* __restrict__ ws_f) {
  const float fx = Kp[0], fy = Kp[4], cx = Kp[2], cy = Kp[5];
  const int* mm = (const int*)(ws_f + WS_MINMAX);
  const float dmin = __int_as_float(mm[0]);
  const float dmax = __int_as_float(mm[1]);
  float4* vn = (float4*)(ws_f + WS_VN);
  for (int pix = blockIdx.x * blockDim.x + threadIdx.x; pix < NPIX;
       pix += gridDim.x * blockDim.x) {
    int py = pix / W_IMG;
    int px = pix - py * W_IMG;
    F3 v = vert1(depth1, py, px, fx, fy, cx, cy);
    F3 n = normal1At(depth1, py, px, fx, fy, cx, cy, dmin, dmax);
    float4 a; a.x = v.x; a.y = v.y; a.z = v.z; a.w = 0.f;
    float4 b; b.x = n.x; b.y = n.y; b.z = n.z; b.w = 0.f;
    vn[pix * 2 + 0] = a;
    vn[pix * 2 + 1] = b;
  }
}

// Main per-iteration kernel: per-pixel ICP + RGB residuals/Jacobians, then the
// 7x7 Gram accumulation JᵀJ / Jᵀr runs on the f32 WMMA pipe (16x16x4).
template <bool PRECOMP>
__global__ __launch_bounds__(128) void icp_accum_kernel(
    const float* __restrict__ depth0, const float* __restrict__ depth1,
    const float* __restrict__ x0, const float* __restrict__ x1,
    const float* __restrict__ Kp, float* __restrict__ ws_f) {
  // 4 waves * 128 augmented-row slices * 16 floats = 32 KB LDS
  __shared__ float smem[4][128 * 16];

  const int tid = threadIdx.x;
  const int wave = tid >> 5;
  const int lane = tid & 31;

  const float fx = Kp[0], fy = Kp[4], cx = Kp[2], cy = Kp[5];
  const float* P = ws_f + WS_POSE;
  const float R00 = P[0],  R01 = P[1],  R02 = P[2],  T0 = P[3];
  const float R10 = P[4],  R11 = P[5],  R12 = P[6],  T1 = P[7];
  const float R20 = P[8],  R21 = P[9],  R22 = P[10], T2 = P[11];
  const int* mm = (const int*)(ws_f + WS_MINMAX);
  const float dmin = __int_as_float(mm[0]);
  const float dmax = __int_as_float(mm[1]);
  const float4* vn = (const float4*)(ws_f + WS_VN);

  v8f acc0 = {0.f, 0.f, 0.f, 0.f, 0.f, 0.f, 0.f, 0.f};
  v8f acc1 = {0.f, 0.f, 0.f, 0.f, 0.f, 0.f, 0.f, 0.f};

  for (int base = blockIdx.x * 128; base < NPIX; base += gridDim.x * 128) {
    int pix = base + tid;
    float av0[7], av1[7], av2[7], av3[7];
#pragma unroll
    for (int k = 0; k < 7; ++k) { av0[k] = 0.f; av1[k] = 0.f; av2[k] = 0.f; av3[k] = 0.f; }

    if (pix < NPIX) {
      int py = pix / W_IMG;
      int px = pix - py * W_IMG;
      float d0 = depth0[pix];
      bool m0 = d0 > 0.0f;
      float vx = ((float)px - cx) / fx * d0;
      float vy = ((float)py - cy) / fy * d0;
      F3 v0t = f3(R00 * vx + R01 * vy + R02 * d0 + T0,
                  R10 * vx + R11 * vy + R12 * d0 + T1,
                  R20 * vx + R21 * vy + R22 * d0 + T2);
      float z = v0t.z;
      float uu = v0t.x / z * fx + cx;
      float vv = v0t.y / z * fy + cy;
      bool inview = (uu > 0.f) && (uu < (float)(W_IMG - 1)) &&
                    (vv > 0.f) && (vv < (float)(H_IMG - 1)) && (z > 0.f);
      // bilinear coords (clipped as in reference)
      float ucl = fminf(fmaxf(uu, 0.f), (float)(W_IMG - 1));
      float vcl = fminf(fmaxf(vv, 0.f), (float)(H_IMG - 1));
      float u0f = floorf(ucl), v0f = floorf(vcl);
      float wu = ucl - u0f, wv = vcl - v0f;
      int u0i = (int)u0f, v0i = (int)v0f;
      int u1i = imin2(u0i + 1, W_IMG - 1);
      int v1i = imin2(v0i + 1, H_IMG - 1);
      float w00 = (1.f - wu) * (1.f - wv), w01 = wu * (1.f - wv);
      float w10 = (1.f - wu) * wv,         w11 = wu * wv;

      F3 v1w, n1w;
      if (PRECOMP) {
        int p00 = v0i * W_IMG + u0i, p01 = v0i * W_IMG + u1i;
        int p10 = v1i * W_IMG + u0i, p11 = v1i * W_IMG + u1i;
        float4 va00 = vn[p00 * 2],     vn00 = vn[p00 * 2 + 1];
        float4 va01 = vn[p01 * 2],     vn01 = vn[p01 * 2 + 1];
        float4 va10 = vn[p10 * 2],     vn10 = vn[p10 * 2 + 1];
        float4 va11 = vn[p11 * 2],     vn11 = vn[p11 * 2 + 1];
        v1w = f3(w00*va00.x + w01*va01.x + w10*va10.x + w11*va11.x,
                 w00*va00.y + w01*va01.y + w10*va10.y + w11*va11.y,
                 w00*va00.z + w01*va01.z + w10*va10.z + w11*va11.z);
        n1w = f3(w00*vn00.x + w01*vn01.x + w10*vn10.x + w11*vn11.x,
                 w00*vn00.y + w01*vn01.y + w10*vn10.y + w11*vn11.y,
                 w00*vn00.z + w01*vn01.z + w10*vn10.z + w11*vn11.z);
      } else {
        F3 a00 = vert1(depth1, v0i, u0i, fx, fy, cx, cy);
        F3 a01 = vert1(depth1, v0i, u1i, fx, fy, cx, cy);
        F3 a10 = vert1(depth1, v1i, u0i, fx, fy, cx, cy);
        F3 a11 = vert1(depth1, v1i, u1i, fx, fy, cx, cy);
        v1w = add3(add3(scl3(a00, w00), scl3(a01, w01)),
                   add3(scl3(a10, w10), scl3(a11, w11)));
        F3 n00 = normal1At(depth1, v0i, u0i, fx, fy, cx, cy, dmin, dmax);
        F3 n01 = normal1At(depth1, v0i, u1i, fx, fy, cx, cy, dmin, dmax);
        F3 n10 = normal1At(depth1, v1i, u0i, fx, fy, cx, cy, dmin, dmax);
        F3 n11 = normal1At(depth1, v1i, u1i, fx, fy, cx, cy, dmin, dmax);
        n1w = add3(add3(scl3(n00, w00), scl3(n01, w01)),
                   add3(scl3(n10, w10), scl3(n11, w11)));
      }
      // bilinear x1
      float x1w[3];
#pragma unroll
      for (int c = 0; c < 3; ++c) {
        x1w[c] = w00 * x1[(v0i * W_IMG + u0i) * 3 + c] +
                 w01 * x1[(v0i * W_IMG + u1i) * 3 + c] +
                 w10 * x1[(v1i * W_IMG + u0i) * 3 + c] +
                 w11 * x1[(v1i * W_IMG + u1i) * 3 + c];
      }

      // ---- ICP residual (point-to-plane, Huber) ----
      F3 diff = sub3(v0t, v1w);
      float dn = sqrtf(dot3(diff, diff));
      bool valid = inview && m0 && (v1w.z > 0.f) && (dn < DIST_THRESH_C);
      float res = valid ? dot3(n1w, diff) : 0.f;
      F3 cr = cross3(v0t, n1w);
      float Ji[6];
      Ji[0] = valid ? cr.x : 0.f;  Ji[1] = valid ? cr.y : 0.f;  Ji[2] = valid ? cr.z : 0.f;
      Ji[3] = valid ? n1w.x : 0.f; Ji[4] = valid ? n1w.y : 0.f; Ji[5] = valid ? n1w.z : 0.f;
      float ax = fabsf(res);
      float rho = (ax <= HUBER_BC) ? ax * ax : 2.f * HUBER_BC * ax - HUBER_BC * HUBER_BC;
      float xs = (ax < 1e-8f) ? 1.f : ax;
      float w = sqrtf(rho + 1e-16f) / xs;
#pragma unroll
      for (int k = 0; k < 6; ++k) av0[k] = w * Ji[k];
      av0[6] = w * res;

      // ---- RGB residuals (3 channels) ----
      bool invR = (!inview) || (!m0);
      float invD = 1.f / d0, invD2 = invD * invD, xyv = vx * vy;
      float Jx6[6], Jy6[6];
      Jx6[0] = fx * (-invD2 * xyv);          Jy6[0] = fy * (-1.f - invD2 * vy * vy);
      Jx6[1] = fx * (1.f + vx * vx * invD2); Jy6[1] = fy * (xyv * invD2);
      Jx6[2] = fx * (-vy * invD);            Jy6[2] = fy * (vx * invD);
      Jx6[3] = fx * invD;                    Jy6[3] = 0.f;
      Jx6[4] = 0.f;                          Jy6[4] = fy * invD;
      Jx6[5] = fx * (-invD2 * vx);           Jy6[5] = fy * (-invD2 * vy);

      int xm = imax2(px - 1, 0), xp = imin2(px + 1, W_IMG - 1);
      int ym = imax2(py - 1, 0), yp = imin2(py + 1, H_IMG - 1);
#pragma unroll
      for (int c = 0; c < 3; ++c) {
        float pmm = x0[(ym * W_IMG + xm) * 3 + c];
        float pm0 = x0[(ym * W_IMG + px) * 3 + c];
        float pmp = x0[(ym * W_IMG + xp) * 3 + c];
        float p0m = x0[(py * W_IMG + xm) * 3 + c];
        float p0p = x0[(py * W_IMG + xp) * 3 + c];
        float ppm = x0[(yp * W_IMG + xm) * 3 + c];
        float pp0 = x0[(yp * W_IMG + px) * 3 + c];
        float ppp = x0[(yp * W_IMG + xp) * 3 + c];
        float gdx = (pmp - pmm) + 2.f * (p0p - p0m) + (ppp - ppm);
        float gdy = (ppm - pmm) + 2.f * (pp0 - pm0) + (ppp - pmp);
        float mag = sqrtf(gdx * gdx + gdy * gdy + 1e-8f);
        float gx = gdx / mag, gy = gdy / mag;
        float resc = invR ? 0.f : (x1w[c] - x0[(py * W_IMG + px) * 3 + c]);
        float* avc = (c == 0) ? av1 : ((c == 1) ? av2 : av3);
#pragma unroll
        for (int k = 0; k < 6; ++k)
          avc[k] = invR ? 0.f : SQRT_LAMBDA_RGB * (gx * Jx6[k] + gy * Jy6[k]);
        avc[6] = SQRT_LAMBDA_RGB * resc;
      }
    }

    __syncthreads(); // previous round's WMMA reads done before we overwrite LDS
    // Stage: slice s = lane*4 + vec, 16 floats per slice (comps 7..15 zero)
    float* sm = smem[wave];
#pragma unroll
    for (int vcn = 0; vcn < 4; ++vcn) {
      const float* src = (vcn == 0) ? av0 : ((vcn == 1) ? av1 : ((vcn == 2) ? av2 : av3));
      int boff = (lane * 4 + vcn) * 16;
#pragma unroll
      for (int k = 0; k < 7; ++k) sm[boff + k] = src[k];
#pragma unroll
      for (int k = 7; k < 16; ++k) sm[boff + k] = 0.f;
    }
    __syncthreads();

    // 32 rank-4 Gram updates: D += A * Aᵀ via V_WMMA_F32_16X16X4_F32.
    // A (16x4) fragment: lane m=lane&15, k = 2*(lane>>4)+vgpr  — identical to
    // the B (4x16) fragment for the symmetric product, so pass frag twice.
    // Two interleaved accumulators break the D->C dependency chain.
    const int mrow = lane & 15;
    const int kh = (lane >> 4) * 2;
#pragma unroll 4
    for (int ch = 0; ch < 32; ++ch) {
      v2f frag;
      frag.x = sm[(ch * 4 + kh) * 16 + mrow];
      frag.y = sm[(ch * 4 + kh + 1) * 16 + mrow];
      if (ch & 1)
        acc1 = __builtin_amdgcn_wmma_f32_16x16x4_f32(
            false, frag, false, frag, (short)0, acc1, false, false);
      else
        acc0 = __builtin_amdgcn_wmma_f32_16x16x4_f32(
            false, frag, false, frag, (short)0, acc0, false, false);
    }
  }

  v8f acc = acc0 + acc1;
  // D layout: lane l (0..15) holds N=l, VGPR g holds M=g. Flush 7x7 corner.
  if (lane < 7) {
#pragma unroll
    for (int g = 0; g < 7; ++g)
      atomicAdd(&ws_f[WS_M + g * 7 + lane], acc[g]);
  }
}

// Single-thread 6x6 damped solve + SE(3) update (tiny serial tail).
__global__ void solve_kernel(float* __restrict__ ws_f, float* __restrict__ out, int write_out) {
  if (threadIdx.x != 0 || blockIdx.x != 0) return;
  float M[49];
  for (int i = 0; i < 49; ++i) M[i] = ws_f[WS_M + i];
  float tr = 0.f;
  for (int i = 0; i < 6; ++i) tr += M[i * 7 + i];
  float lam = tr * DAMPING_C;
  float A[6][7];
  for (int i = 0; i < 6; ++i) {
    for (int j = 0; j < 6; ++j) A[i][j] = M[i * 7 + j] + ((i == j) ? lam : 0.f);
    A[i][6] = M[i * 7 + 6];
  }
  // Gauss-Jordan with partial pivoting
  for (int k = 0; k < 6; ++k) {
    int p = k;
    float best = fabsf(A[k][k]);
    for (int r = k + 1; r < 6; ++r) { float v = fabsf(A[r][k]); if (v > best) { best = v; p = r; } }
    if (p != k) for (int c = k; c < 7; ++c) { float t = A[k][c]; A[k][c] = A[p][c]; A[p][c] = t; }
    float ip = 1.f / A[k][k];
    for (int c = k; c < 7; ++c) A[k][c] *= ip;
    for (int r = 0; r < 6; ++r) {
      if (r == k) continue;
      float f = A[r][k];
      for (int c = k; c < 7; ++c) A[r][c] -= f * A[k][c];
    }
  }
  float xi[6];
  for (int i = 0; i < 6; ++i) xi[i] = A[i][6];

  // dR = exp_so3(-xi[0:3])
  float wx = -xi[0], wy = -xi[1], wz = -xi[2];
  float th2 = fmaxf(wx * wx + wy * wy + wz * wz, 1e-30f);
  float th = sqrtf(th2);
  float sA = sinf(th) / th;
  float sB = (1.f - cosf(th)) / th2;
  float Wm[3][3] = {{0.f, -wz, wy}, {wz, 0.f, -wx}, {-wy, wx, 0.f}};
  float W2[3][3];
  for (int i = 0; i < 3; ++i)
    for (int j = 0; j < 3; ++j) {
      float s = 0.f;
      for (int k = 0; k < 3; ++k) s += Wm[i][k] * Wm[k][j];
      W2[i][j] = s;
    }
  float dR[3][3];
  bool ident = (th <= 1e-10f);
  for (int i = 0; i < 3; ++i)
    for (int j = 0; j < 3; ++j) {
      float v = ((i == j) ? 1.f : 0.f) + sA * Wm[i][j] + sB * W2[i][j];
      dR[i][j] = ident ? ((i == j) ? 1.f : 0.f) : v;
    }
  float dt[3];
  for (int i = 0; i < 3; ++i)
    dt[i] = -(dR[i][0] * xi[3] + dR[i][1] * xi[4] + dR[i][2] * xi[5]);

  // pose update: R1 = dR @ Rp ; t1 = dR @ tp + dt
  float Pp[16];
  for (int i = 0; i < 16; ++i) Pp[i] = ws_f[WS_POSE + i];
  float Pn[16];
  for (int i = 0; i < 3; ++i) {
    for (int j = 0; j < 3; ++j) {
      float s = 0.f;
      for (int k = 0; k < 3; ++k) s += dR[i][k] * Pp[k * 4 + j];
      Pn[i * 4 + j] = s;
    }
    float s = 0.f;
    for (int k = 0; k < 3; ++k) s += dR[i][k] * Pp[k * 4 + 3];
    Pn[i * 4 + 3] = s + dt[i];
  }
  Pn[12] = 0.f; Pn[13] = 0.f; Pn[14] = 0.f; Pn[15] = 1.f;
  for (int i = 0; i < 16; ++i) ws_f[WS_POSE + i] = Pn[i];
  if (write_out) for (int i = 0; i < 16; ++i) out[i] = Pn[i];
  // re-zero accumulator for next iteration
  for (int i = 0; i < 49; ++i) ws_f[WS_M + i] = 0.f;
}

extern "C" void kernel_launch(void* const* d_in, const int* in_sizes, int n_in,
                              void* d_out, int out_size, void* d_ws, size_t ws_size,
                              hipStream_t stream) {
  (void)in_sizes; (void)n_in; (void)out_size;
  const float* pose10 = (const float*)d_in[0];
  const float* depth0 = (const float*)d_in[1];
  const float* depth1 = (const float*)d_in[2];
  const float* x0     = (const float*)d_in[3];
  const float* x1     = (const float*)d_in[4];
  const float* Kp     = (const float*)d_in[5];
  float* ws_f = (float*)d_ws;
  float* out  = (float*)d_out;

  const bool big_ws = (ws_size >= WS_VN_BYTES);

  init_kernel<<<1, 64, 0, stream>>>(pose10, ws_f);
  minmax_kernel<<<256, 256, 0, stream>>>(depth1, ws_f);
  if (big_ws) {
    precompute_vn_kernel<<<1920, 256, 0, stream>>>(depth1, Kp, ws_f);
    for (int it = 0; it < 3; ++it) {
      icp_accum_kernel<true><<<960, 128, 0, stream>>>(depth0, depth1, x0, x1, Kp, ws_f);
      solve_kernel<<<1, 1, 0, stream>>>(ws_f, out, (it == 2) ? 1 : 0);
    }
  } else {
    for (int it = 0; it < 3; ++it) {
      icp_accum_kernel<false><<<960, 128, 0, stream>>>(depth0, depth1, x0, x1, Kp, ws_f);
      solve_kernel<<<1, 1, 0, stream>>>(ws_f, out, (it == 2) ? 1 : 0);
    }
  }
}